// LightGCN_30399778521336
// MI455X (gfx1250) — compile-verified
//
#include <hip/hip_runtime.h>
#include <cstdint>
#include <cstddef>

// ---------------------------------------------------------------------------
// LightGCN propagation for MI455X (gfx1250).
//   acc = (E + A E + A^2 E + A^3 E) / 4
// Memory-bound SpMM (AI ~0.25 flop/byte) -> no WMMA; design targets:
//   * 192MB L2: both 76.8MB embedding buffers stay L2-resident, so the
//     random 256B row gathers and fp32 scatter atomics resolve in L2.
//   * TDM (tensor_load_to_lds) stages the sequential edge stream into LDS.
//   * wave32: 16 lanes x float4 = one 256B row, 2 edges per wave-iteration.
// ---------------------------------------------------------------------------

#define N_USERS   200000
#define N_ITEMS   100000
#define N_NODES   300000
#define DIM       64
#define NNZ       2000000
#define N_LAYERS  3

#define EMB_ELEMS ((size_t)N_NODES * DIM)          // 19,200,000 floats (76.8 MB)
#define U_ELEMS   ((size_t)N_USERS * DIM)          // 12,800,000 floats

#define BLOCK  256
#define WAVES  (BLOCK / 32)
#define CHUNK  2048                                 // edges staged per block
#define EDGES_PER_WAVE (CHUNK / WAVES)              // 256

typedef __attribute__((ext_vector_type(4))) unsigned int v4u;
typedef __attribute__((ext_vector_type(8))) int          v8i;
typedef __attribute__((ext_vector_type(4))) int          v4i;

#if defined(__has_builtin)
#  if __has_builtin(__builtin_amdgcn_tensor_load_to_lds) && \
      __has_builtin(__builtin_amdgcn_s_wait_tensorcnt)
#    define USE_TDM 1
#  endif
#endif
#ifndef USE_TDM
#  define USE_TDM 0
#endif

#if USE_TDM
// Issue one TDM 1-D copy: `tile_elems` 4-byte elements from global -> LDS.
// Descriptor per CDNA5 ISA ch.8: 2D tensor, one row (tile_dim1 = 1),
// tensor_dim0 = valid elements remaining (tail chunk zero-fills OOB).
// This toolchain's builtin is the 6-arg form: (v4u, v8i, v4i, v4i, v8i, cpol).
__device__ __forceinline__ void tdm_load_1d(unsigned lds_off, const void* gptr,
                                            unsigned elems_valid,
                                            unsigned tile_elems) {
  unsigned long long ga = (unsigned long long)(uintptr_t)gptr;
  v4u g0;
  g0[0] = 1u;                                   // count=1, user descriptor
  g0[1] = lds_off;                              // lds_addr (bytes)
  g0[2] = (unsigned)ga;                         // global_addr[31:0]
  g0[3] = ((unsigned)(ga >> 32) & 0x01FFFFFFu)  // global_addr[56:32]
          | 0x80000000u;                        // type = 2 ("image")
  v8i g1;
  g1[0] = (int)(2u << 16);                      // data_size = 4 bytes
  g1[1] = (int)((elems_valid & 0xFFFFu) << 16); // tensor_dim0[15:0]
  g1[2] = (int)(((elems_valid >> 16) & 0xFFFFu) // tensor_dim0[31:16]
          | (1u << 16));                        // tensor_dim1 = 1
  g1[3] = (int)(tile_elems << 16);              // tile_dim0
  g1[4] = 1;                                    // tile_dim1 = 1, tile_dim2 = 0
  g1[5] = 0;                                    // dim0_stride (unused: 1 row)
  g1[6] = 0;
  g1[7] = 0;
  v4i z4 = {0, 0, 0, 0};                        // groups 2/3 unused (<=2D)
  v8i z8 = {0, 0, 0, 0, 0, 0, 0, 0};
  __builtin_amdgcn_tensor_load_to_lds(g0, g1, z4, z4, z8, 0);
}
#endif

// ---------------------------------------------------------------------------
// scatter: y[dst] += val * x[src]  (one layer of SpMM). Hot kernel.
// Edge stream staged into LDS via TDM; 2 edges per wave-iteration,
// 16 lanes x float4 per edge row; non-returning global fp32 atomics.
// (Defined first so the disasm snippet shows this kernel.)
// ---------------------------------------------------------------------------
__global__ void __launch_bounds__(BLOCK) lgcn_scatter(
    const float* __restrict__ x, float* __restrict__ y,
    const int* __restrict__ esrc, const int* __restrict__ edst,
    const float* __restrict__ eval) {
  __shared__ int   s_src[CHUNK];
  __shared__ int   s_dst[CHUNK];
  __shared__ float s_val[CHUNK];

  const int chunkStart = (int)blockIdx.x * CHUNK;
  const int remain = NNZ - chunkStart;            // > 0 by construction

#if USE_TDM
  if (threadIdx.x < 32) {                         // one wave issues the DMAs
    const unsigned n = (unsigned)remain;
    tdm_load_1d((unsigned)(size_t)&s_src[0], esrc + chunkStart, n, CHUNK);
    tdm_load_1d((unsigned)(size_t)&s_dst[0], edst + chunkStart, n, CHUNK);
    tdm_load_1d((unsigned)(size_t)&s_val[0], eval + chunkStart, n, CHUNK);
    __builtin_amdgcn_s_wait_tensorcnt(0);
  }
#else
  for (int t = (int)threadIdx.x; t < CHUNK; t += BLOCK) {
    const int g = chunkStart + t;
    const bool ok = (g < NNZ);
    s_src[t] = ok ? esrc[g] : 0;
    s_dst[t] = ok ? edst[g] : 0;
    s_val[t] = ok ? eval[g] : 0.f;
  }
#endif
  __syncthreads();

  const int lane = (int)(threadIdx.x & 31);
  const int wave = (int)(threadIdx.x >> 5);
  const int half = lane >> 4;                     // which of the 2 edges
  const int col  = (lane & 15) * 4;               // 16 lanes x 4 floats = row
  const int base = wave * EDGES_PER_WAVE;

#pragma unroll 4
  for (int i = 0; i < EDGES_PER_WAVE; i += 2) {
    const int e = base + i + half;
    if (e < remain) {
      const int   s = s_src[e];
      const int   d = s_dst[e];
      const float w = s_val[e];
      const float4 v = *(const float4*)(x + (size_t)s * DIM + col);
      float* o = y + (size_t)d * DIM + col;
      unsafeAtomicAdd(o + 0, w * v.x);
      unsafeAtomicAdd(o + 1, w * v.y);
      unsafeAtomicAdd(o + 2, w * v.z);
      unsafeAtomicAdd(o + 3, w * v.w);
    }
  }
}

// ---------------------------------------------------------------------------
// init: embA = concat(user,item); embB = 0; acc = 0.25 * emb
// ---------------------------------------------------------------------------
__global__ void __launch_bounds__(BLOCK) lgcn_init(
    const float* __restrict__ user, const float* __restrict__ item,
    float* __restrict__ embA, float* __restrict__ embB,
    float* __restrict__ acc) {
  const size_t i = ((size_t)blockIdx.x * BLOCK + threadIdx.x) * 4;
  if (i >= EMB_ELEMS) return;
  const float4 v = (i < U_ELEMS) ? *(const float4*)(user + i)
                                 : *(const float4*)(item + (i - U_ELEMS));
  *(float4*)(embA + i) = v;
  float4 zz; zz.x = 0.f; zz.y = 0.f; zz.z = 0.f; zz.w = 0.f;
  *(float4*)(embB + i) = zz;
  float4 a;
  a.x = 0.25f * v.x; a.y = 0.25f * v.y; a.z = 0.25f * v.z; a.w = 0.25f * v.w;
  *(float4*)(acc + i) = a;
}

// ---------------------------------------------------------------------------
// accum: acc += 0.25 * tmp; optionally zero the dead ping-pong buffer (fused
// so the next layer's scatter target is clean without an extra pass). The
// final layer skips the zeroing (no next scatter -> saves a 77MB write pass).
// ---------------------------------------------------------------------------
template <bool ZERO_DEAD>
__global__ void __launch_bounds__(BLOCK) lgcn_accum(
    const float* __restrict__ tmp, float* __restrict__ acc,
    float* __restrict__ dead) {
  const size_t i = ((size_t)blockIdx.x * BLOCK + threadIdx.x) * 4;
  if (i >= EMB_ELEMS) return;
  const float4 t = *(const float4*)(tmp + i);
  float4 a = *(float4*)(acc + i);
  a.x += 0.25f * t.x; a.y += 0.25f * t.y;
  a.z += 0.25f * t.z; a.w += 0.25f * t.w;
  *(float4*)(acc + i) = a;
  if (ZERO_DEAD) {
    float4 zz; zz.x = 0.f; zz.y = 0.f; zz.z = 0.f; zz.w = 0.f;
    *(float4*)(dead + i) = zz;
  }
}

// ---------------------------------------------------------------------------
extern "C" void kernel_launch(void* const* d_in, const int* in_sizes, int n_in,
                              void* d_out, int out_size, void* d_ws,
                              size_t ws_size, hipStream_t stream) {
  (void)in_sizes; (void)n_in; (void)out_size; (void)ws_size;
  const float* user = (const float*)d_in[0];
  const float* item = (const float*)d_in[1];
  const float* eval = (const float*)d_in[2];
  const int*   esrc = (const int*)d_in[3];
  const int*   edst = (const int*)d_in[4];

  float* acc  = (float*)d_out;                    // accumulated output (node order)
  float* embA = (float*)d_ws;                     // 76.8 MB
  float* embB = embA + EMB_ELEMS;                 // 76.8 MB (both L2-resident)

  const int denseBlocks   = (int)((EMB_ELEMS / 4 + BLOCK - 1) / BLOCK); // 18750
  const int scatterBlocks = (NNZ + CHUNK - 1) / CHUNK;                  // 977

  lgcn_init<<<denseBlocks, BLOCK, 0, stream>>>(user, item, embA, embB, acc);

  float* a = embA;
  float* b = embB;
  for (int l = 0; l < N_LAYERS; ++l) {
    lgcn_scatter<<<scatterBlocks, BLOCK, 0, stream>>>(a, b, esrc, edst, eval);
    if (l + 1 < N_LAYERS) {
      lgcn_accum<true><<<denseBlocks, BLOCK, 0, stream>>>(b, acc, a);
    } else {
      lgcn_accum<false><<<denseBlocks, BLOCK, 0, stream>>>(b, acc, a);
    }
    float* t = a; a = b; b = t;                   // ping-pong
  }
}